// Loss_90494960926896
// MI455X (gfx1250) — compile-verified
//
#include <hip/hip_runtime.h>
#include <math.h>

#define NB 512
#define NV 778
#define NF 1538
#define NP 1024
#define NV2 (2 * NV)
#define NF2 (2 * NF)

typedef __attribute__((ext_vector_type(2))) float v2f;
typedef __attribute__((ext_vector_type(8))) float v8f;

// ---------------------------------------------------------------------------
// Wave-level sum via V_WMMA_F32_16X16X4_F32:
// A[m,k]: lane L<16 holds (A[L,0],A[L,1]) in VGPR0/1, lane 16+L holds (A[L,2],A[L,3]).
// We place each lane's partial in VGPR0 and 0 in VGPR1; B = all ones (layout-
// independent), C = 0.  D[m,n] = rowsum(m) = x(m) + x(m+16).  Lane 0 holds
// D[0..7,0] in its 8 D regs, lane 16 holds D[8..15,0]; their sums add to the
// full 32-lane total.  Exact f32 reduction, one matrix op.
// ---------------------------------------------------------------------------
__device__ __forceinline__ float wave_sum_wmma(float x) {
  v2f a = {x, 0.0f};
  v2f b = {1.0f, 1.0f};
  v8f c = {};
  v8f d = __builtin_amdgcn_wmma_f32_16x16x4_f32(
      /*neg_a=*/false, a, /*neg_b=*/false, b,
      /*c_mod=*/(short)0, c, /*reuse_a=*/false, /*reuse_b=*/false);
  float y = d[0] + d[1] + d[2] + d[3] + d[4] + d[5] + d[6] + d[7];
  return __shfl(y, 0, 32) + __shfl(y, 16, 32);
}

struct F3 { float x, y, z; };

__device__ __forceinline__ F3 ldv(const float* s, int v) {
  F3 r; r.x = s[3 * v + 0]; r.y = s[3 * v + 1]; r.z = s[3 * v + 2]; return r;
}

// phi.sum for one triangle (a,b,c) against 3 points; SIGMA=0.5 -> exp(-2*rho2)
__device__ __forceinline__ float cone3(F3 a, F3 b, F3 c, F3 p0, F3 p1, F3 p2) {
  const float third = 1.0f / 3.0f;
  float cx = (a.x + b.x + c.x) * third;
  float cy = (a.y + b.y + c.y) * third;
  float cz = (a.z + b.z + c.z) * third;
  float e1x = b.x - a.x, e1y = b.y - a.y, e1z = b.z - a.z;
  float e2x = c.x - a.x, e2y = c.y - a.y, e2z = c.z - a.z;
  float nx = e1y * e2z - e1z * e2y;
  float ny = e1z * e2x - e1x * e2z;
  float nz = e1x * e2y - e1y * e2x;
  float inv = 1.0f / (sqrtf(nx * nx + ny * ny + nz * nz) + 1e-9f);
  nx *= inv; ny *= inv; nz *= inv;
  float acc = 0.0f;
  {
    float dx = p0.x - cx, dy = p0.y - cy, dz = p0.z - cz;
    float h = dx * nx + dy * ny + dz * nz;
    float d2 = dx * dx + dy * dy + dz * dz;
    float rho2 = fmaxf(d2 - h * h, 0.0f);
    acc += fmaxf(-h, 0.0f) * expf(-2.0f * rho2);
  }
  {
    float dx = p1.x - cx, dy = p1.y - cy, dz = p1.z - cz;
    float h = dx * nx + dy * ny + dz * nz;
    float d2 = dx * dx + dy * dy + dz * dz;
    float rho2 = fmaxf(d2 - h * h, 0.0f);
    acc += fmaxf(-h, 0.0f) * expf(-2.0f * rho2);
  }
  {
    float dx = p2.x - cx, dy = p2.y - cy, dz = p2.z - cz;
    float h = dx * nx + dy * ny + dz * nz;
    float d2 = dx * dx + dy * dy + dz * dz;
    float rho2 = fmaxf(d2 - h * h, 0.0f);
    acc += fmaxf(-h, 0.0f) * expf(-2.0f * rho2);
  }
  return acc;
}

// ---------------------------------------------------------------------------
// Kernel A: per-batch collision loss.  One block per batch; verts (both hands)
// and the remapped combined face table staged in LDS (55.6 KB / 320 KB WGP).
// Per-wave WMMA reduction, per-batch result -> ws[b].
// ---------------------------------------------------------------------------
__global__ __launch_bounds__(256) void collision_kernel(
    const float* __restrict__ ov, const int* __restrict__ faces,
    const int* __restrict__ cidx, float* __restrict__ ws) {
  __shared__ float s_verts[NV2 * 3];
  __shared__ int s_faces[NF2 * 3];
  __shared__ float s_red[8];

  const int b = blockIdx.x;
  const int tid = threadIdx.x;

  const float* v0 = ov + (size_t)b * NV * 3;              // hand 0
  const float* v1 = ov + (size_t)(NB + b) * NV * 3;       // hand 1
  for (int v = tid; v < NV; v += 256) {
    s_verts[3 * v + 0] = v0[3 * v + 0];
    s_verts[3 * v + 1] = v0[3 * v + 1];
    s_verts[3 * v + 2] = v0[3 * v + 2];
    s_verts[3 * (v + NV) + 0] = v1[3 * v + 0];
    s_verts[3 * (v + NV) + 1] = v1[3 * v + 1];
    s_verts[3 * (v + NV) + 2] = v1[3 * v + 2];
  }
  for (int f = tid; f < NF; f += 256) {
    s_faces[3 * f + 0] = faces[3 * f + 0];
    s_faces[3 * f + 1] = faces[3 * f + 1];
    s_faces[3 * f + 2] = faces[3 * f + 2];
    s_faces[3 * (f + NF) + 0] = faces[3 * (NF + f) + 0] + NV;
    s_faces[3 * (f + NF) + 1] = faces[3 * (NF + f) + 1] + NV;
    s_faces[3 * (f + NF) + 2] = faces[3 * (NF + f) + 2] + NV;
  }
  __syncthreads();

  float s = 0.0f;
  const int* cb = cidx + (size_t)b * NP * 2;
#pragma unroll
  for (int it = 0; it < NP / 256; ++it) {
    int p = tid + it * 256;
    int c0 = cb[2 * p + 0];
    int c1 = cb[2 * p + 1];
    bool vld = (c0 >= 0) && (c1 >= 0);
    int f0 = c0 < 0 ? 0 : c0;
    int f1 = c1 < 0 ? 0 : c1;
    int r0 = s_faces[3 * f0 + 0], r1 = s_faces[3 * f0 + 1], r2 = s_faces[3 * f0 + 2];
    int i0 = s_faces[3 * f1 + 0], i1 = s_faces[3 * f1 + 1], i2 = s_faces[3 * f1 + 2];
    F3 ra = ldv(s_verts, r0), rb = ldv(s_verts, r1), rc = ldv(s_verts, r2);
    F3 ia = ldv(s_verts, i0), ib = ldv(s_verts, i1), ic = ldv(s_verts, i2);
    float pr = cone3(ra, rb, rc, ia, ib, ic) + cone3(ia, ib, ic, ra, rb, rc);
    s += vld ? pr : 0.0f;   // select, not branch: EXEC stays all-ones for WMMA
  }

  float wsum = wave_sum_wmma(s);
  int lane = tid & 31, wid = tid >> 5;
  if (lane == 0) s_red[wid] = wsum;
  __syncthreads();
  if (tid == 0) {
    float t = 0.0f;
#pragma unroll
    for (int i = 0; i < 8; ++i) t += s_red[i];
    ws[b] = t;
  }
}

// ---------------------------------------------------------------------------
// Generic block reduction (valid on thread 0)
// ---------------------------------------------------------------------------
__device__ float block_sum(float v, float* sb) {
  int lane = threadIdx.x & 31, wid = threadIdx.x >> 5;
#pragma unroll
  for (int o = 16; o; o >>= 1) v += __shfl_down(v, o, 32);
  __syncthreads();
  if (lane == 0) sb[wid] = v;
  __syncthreads();
  float r = (threadIdx.x < 8) ? sb[threadIdx.x] : 0.0f;
  if (wid == 0) {
#pragma unroll
    for (int o = 4; o; o >>= 1) r += __shfl_down(r, o, 32);
  }
  return r;
}

// ---------------------------------------------------------------------------
// Kernel B: all scalar losses.  11 blocks, each computes one output term.
//  comp 0..2  -> out[1..3]  (inter-hand masked means)
//  comp 3..8  -> out[4..9]  (per-hand masked means, summed over h)
//  comp 9     -> out[11]    (weighted CE)
//  comp 10    -> out[0]     (collision finalize from ws) and out[10]=0 (reg)
// ---------------------------------------------------------------------------
__global__ __launch_bounds__(256) void losses_kernel(
    const float* __restrict__ betas, const float* __restrict__ transl,
    const float* __restrict__ j3d, const float* __restrict__ go,
    const float* __restrict__ pose, const float* __restrict__ tt,
    const float* __restrict__ tj, const float* __restrict__ tg,
    const float* __restrict__ tp, const float* __restrict__ tsh,
    const float* __restrict__ logits, const int* __restrict__ handed,
    const int* __restrict__ valid, const int* __restrict__ ctgt,
    const float* __restrict__ ws, float* __restrict__ out) {
  __shared__ float sb[8];
  const int tid = threadIdx.x;
  const int comp = blockIdx.x;

  if (comp <= 2) {
    // inter-hand terms
    const int cols = (comp == 0) ? 10 : (comp == 1) ? 3 : 63;
    float num = 0.0f;
    for (int idx = tid; idx < NB * cols; idx += 256) {
      int b = idx / cols, c = idx % cols;
      if (handed[2 * b] + handed[2 * b + 1] == 2) {
        float e;
        if (comp == 0) {
          float d = betas[b * 10 + c] - betas[(NB + b) * 10 + c];
          e = d * d;
        } else if (comp == 1) {
          float d = (transl[b * 3 + c] - transl[(NB + b) * 3 + c]) -
                    (tt[b * 3 + c] - tt[(NB + b) * 3 + c]);
          e = d * d;
        } else {
          float d = (j3d[b * 63 + c] - j3d[(NB + b) * 63 + c]) -
                    (tj[b * 63 + c] - tj[(NB + b) * 63 + c]);
          e = d * d;
        }
        num += e;
      }
    }
    float cnt = 0.0f;
    for (int b = tid; b < NB; b += 256)
      cnt += (handed[2 * b] + handed[2 * b + 1] == 2) ? 1.0f : 0.0f;
    num = block_sum(num, sb);
    cnt = block_sum(cnt, sb);
    if (tid == 0) {
      float scale = (comp == 0) ? 1.0f : 100.0f;
      float den = cnt * (float)cols;
      out[1 + comp] = (den > 0.0f) ? num / den * scale : 0.0f;
    }
  } else if (comp <= 8) {
    // per-hand terms, summed over h with per-h denominators
    int cols, oidx;
    float scale;
    switch (comp) {
      case 3: cols = 3;  scale = 10.0f;  oidx = 4; break;  // lgo
      case 4: cols = 45; scale = 10.0f;  oidx = 5; break;  // lhp
      case 5: cols = 60; scale = 0.01f;  oidx = 6; break;  // lrj
      case 6: cols = 63; scale = 0.01f;  oidx = 7; break;  // lj3
      case 7: cols = 10; scale = 10.0f;  oidx = 8; break;  // lsh
      default: cols = 3; scale = 10.0f;  oidx = 9; break;  // ltr
    }
    float res = 0.0f;
    for (int h = 0; h < 2; ++h) {
      float num = 0.0f;
      for (int idx = tid; idx < NB * cols; idx += 256) {
        int b = idx / cols, c = idx % cols;
        if (valid[h * NB + b]) {
          float e;
          if (comp == 3) {
            float d = go[(h * NB + b) * 3 + c] - tg[(h * NB + b) * 3 + c];
            e = d * d;
          } else if (comp == 4) {
            float d = pose[(h * NB + b) * 45 + c] - tp[(h * NB + b) * 45 + c];
            e = d * d;
          } else if (comp == 5) {
            int jj = c / 3 + 1, cc = c % 3;
            const float* oj = j3d + (size_t)(h * NB + b) * 63;
            const float* tjp = tj + (size_t)(h * NB + b) * 63;
            float a1 = (oj[jj * 3 + cc] - oj[cc]) * 1000.0f;
            float b1 = (tjp[jj * 3 + cc] - tjp[cc]) * 1000.0f;
            e = fabsf(a1 - b1);
          } else if (comp == 6) {
            float a1 = j3d[(size_t)(h * NB + b) * 63 + c] * 1000.0f;
            float b1 = tj[(size_t)(h * NB + b) * 63 + c] * 1000.0f;
            e = fabsf(a1 - b1);
          } else if (comp == 7) {
            float d = betas[(h * NB + b) * 10 + c] - tsh[(h * NB + b) * 10 + c];
            e = d * d;
          } else {
            e = fabsf(transl[(h * NB + b) * 3 + c] - tt[(h * NB + b) * 3 + c]);
          }
          num += e;
        }
      }
      float cnt = 0.0f;
      for (int b = tid; b < NB; b += 256)
        cnt += valid[h * NB + b] ? 1.0f : 0.0f;
      num = block_sum(num, sb);
      cnt = block_sum(cnt, sb);
      if (tid == 0) {
        float den = cnt * (float)cols;
        res += (den > 0.0f) ? num / den * scale : 0.0f;
      }
    }
    if (tid == 0) out[oidx] = res;
  } else if (comp == 9) {
    // weighted cross entropy
    const float cw0 = 1.0f, cw1 = 30.0f, cw2 = 30.0f, cw3 = 10.0f;
    float num = 0.0f, den = 0.0f;
    for (int b = tid; b < NB; b += 256) {
      int t = ctgt[b];
      if (t != 0) {
        float l0 = logits[4 * b + 0], l1 = logits[4 * b + 1];
        float l2 = logits[4 * b + 2], l3 = logits[4 * b + 3];
        float m = fmaxf(fmaxf(l0, l1), fmaxf(l2, l3));
        float z = expf(l0 - m) + expf(l1 - m) + expf(l2 - m) + expf(l3 - m);
        float lt = logits[4 * b + t];
        float nll = -(lt - m - logf(z));
        float w = (t == 1) ? cw1 : (t == 2) ? cw2 : (t == 3) ? cw3 : cw0;
        num += w * nll;
        den += w;
      }
    }
    num = block_sum(num, sb);
    den = block_sum(den, sb);
    if (tid == 0) out[11] = num / fmaxf(den, 1e-9f);
  } else {
    // collision finalize + reg (identically zero: mse(x,x) terms)
    float s = 0.0f, c = 0.0f;
    for (int b = tid; b < NB; b += 256) {
      float lb = ws[b];
      if (lb != 0.0f) { s += lb; c += 1.0f; }
    }
    s = block_sum(s, sb);
    c = block_sum(c, sb);
    if (tid == 0) {
      out[0] = (c > 0.0f) ? s / fmaxf(c, 1.0f) * 100.0f : 0.0f;
      out[10] = 0.0f;
    }
  }
}

extern "C" void kernel_launch(void* const* d_in, const int* in_sizes, int n_in,
                              void* d_out, int out_size, void* d_ws, size_t ws_size,
                              hipStream_t stream) {
  const float* ov      = (const float*)d_in[0];   // (2,B,V,3)
  const float* betas   = (const float*)d_in[1];   // (2,B,10)
  const float* transl  = (const float*)d_in[2];   // (2,B,3)
  const float* j3d     = (const float*)d_in[3];   // (2,B,21,3)
  const float* go      = (const float*)d_in[4];   // (2,B,3)
  const float* pose    = (const float*)d_in[5];   // (2,B,45)
  const float* tt      = (const float*)d_in[6];   // (2,B,3)
  const float* tj      = (const float*)d_in[7];   // (2,B,21,3)
  const float* tg      = (const float*)d_in[8];   // (2,B,3)
  const float* tp      = (const float*)d_in[9];   // (2,B,45)
  const float* tsh     = (const float*)d_in[10];  // (2,B,10)
  const float* logits  = (const float*)d_in[11];  // (B,4)
  const int*   faces   = (const int*)d_in[12];    // (2,F,3)
  const int*   cidx    = (const int*)d_in[13];    // (B,P,2)
  const int*   handed  = (const int*)d_in[14];    // (B,2)
  const int*   valid   = (const int*)d_in[15];    // (2,B)
  const int*   ctgt    = (const int*)d_in[16];    // (B,)
  float* out = (float*)d_out;                     // 12 scalars
  float* ws  = (float*)d_ws;                      // >= NB floats

  collision_kernel<<<NB, 256, 0, stream>>>(ov, faces, cidx, ws);
  losses_kernel<<<11, 256, 0, stream>>>(betas, transl, j3d, go, pose, tt, tj,
                                        tg, tp, tsh, logits, handed, valid,
                                        ctgt, ws, out);
}